// ColorLookup_47974784697158
// MI455X (gfx1250) — compile-verified
//
#include <hip/hip_runtime.h>

typedef __attribute__((ext_vector_type(2))) float v2f;
typedef __attribute__((ext_vector_type(8))) float v8f;

#define NPIX     524288      // 8*256*256
#define CSTRIDE  65536       // H*W
#define BSTRIDE  196608      // 3*H*W
#define NTAB     512
#define NTILE    32          // 512/16
#define NBLOCKS  4096        // NPIX / (8 waves * 16 pixels)

// Each wave handles 16 pixels. Per 16-palette tile:
//   score[m][n] = t2[m] - 2 * dot(pixel_n, table_m)  via V_WMMA_F32_16X16X4_F32
// A (16x4):  lane m (lo): v0=-2r, v1=-2g ; lane m+16 (hi): v0=-2b, v1=||t||^2
// B (4x16):  lane n (lo): v0=x, v1=y     ; lane n+16 (hi): v0=z,  v1=1.0
// D (16x16): lane n, vgpr i -> M = i + 8*half  (half = lane>=16)
__global__ __launch_bounds__(256) void vq_main(const float* __restrict__ zin,
                                               const float* __restrict__ table,
                                               float* __restrict__ outp,
                                               float* __restrict__ partials) {
    __shared__ float ldsA[NTILE * 64];   // [tile][lane][2]  -- A-operand layout
    __shared__ float ldsT[NTAB * 3];     // raw palette for final gather
    __shared__ float wpart[8];

    // Stage palette into LDS in WMMA A-operand layout.
    for (int m = threadIdx.x; m < NTAB; m += 256) {
        float r = table[m * 3 + 0];
        float g = table[m * 3 + 1];
        float b = table[m * 3 + 2];
        int tile = m >> 4, mm = m & 15;
        ldsA[tile * 64 + mm * 2 + 0]        = -2.0f * r;   // K=0 (lanes 0-15, v0)
        ldsA[tile * 64 + mm * 2 + 1]        = -2.0f * g;   // K=1 (lanes 0-15, v1)
        ldsA[tile * 64 + (mm + 16) * 2 + 0] = -2.0f * b;   // K=2 (lanes 16-31, v0)
        ldsA[tile * 64 + (mm + 16) * 2 + 1] = r*r + g*g + b*b; // K=3 (lanes 16-31, v1)
        ldsT[m * 3 + 0] = r;
        ldsT[m * 3 + 1] = g;
        ldsT[m * 3 + 2] = b;
    }
    __syncthreads();

    const int lane = threadIdx.x & 31;
    const int wid  = threadIdx.x >> 5;
    const int n    = lane & 15;
    const int half = lane >> 4;

    const int p    = (blockIdx.x * 8 + wid) * 16 + n;  // pixel index
    const int bi   = p >> 16;                          // image in batch
    const int hw   = p & 65535;
    const int addr = bi * BSTRIDE + hw;                // address of channel 0

    // Build B operand (no divergent branches: unconditional loads + select).
    float b0 = zin[addr + half * (2 * CSTRIDE)];       // lo: x   hi: z
    float yv = zin[addr + CSTRIDE];
    float b1 = half ? 1.0f : yv;                       // lo: y   hi: 1.0
    v2f bvec;
    bvec.x = b0;
    bvec.y = b1;

    float minval = 3.4e38f;
    int   minidx = 0;
    const int idxbase = half * 8;
    v8f czero = {0.f, 0.f, 0.f, 0.f, 0.f, 0.f, 0.f, 0.f};

#pragma unroll
    for (int tile = 0; tile < NTILE; ++tile) {
        v2f a = *(const v2f*)&ldsA[tile * 64 + lane * 2];
        // (neg_a, A, neg_b, B, c_mod, C, reuse_a, reuse_b)
        v8f c = __builtin_amdgcn_wmma_f32_16x16x4_f32(
            false, a, false, bvec, (short)0, czero, false, false);
#pragma unroll
        for (int i = 0; i < 8; ++i) {
            float v  = c[i];
            int  idx = tile * 16 + idxbase + i;
            if (v < minval) { minval = v; minidx = idx; }  // strict '<': lowest-index ties
        }
    }

    // Combine the two palette halves for pixel n (lanes n and n+16).
    float zc   = __shfl_xor(b0, 16, 32);     // lo lanes receive the z component
    float oval = __shfl_xor(minval, 16, 32);
    int   oidx = __shfl_xor(minidx, 16, 32);
    if (oval < minval || (oval == minval && oidx < minidx)) {
        minval = oval;
        minidx = oidx;
    }

    float lsum = 0.0f;
    if (half == 0) {
        float qr = ldsT[minidx * 3 + 0];
        float qg = ldsT[minidx * 3 + 1];
        float qb = ldsT[minidx * 3 + 2];
        outp[addr]               = qr;
        outp[addr + CSTRIDE]     = qg;
        outp[addr + 2 * CSTRIDE] = qb;
        float dx = qr - b0, dy = qg - b1, dz = qb - zc;
        lsum = dx * dx + dy * dy + dz * dz;
    }

    // Deterministic loss partial: wave reduce -> block reduce -> d_ws[block]
#pragma unroll
    for (int off = 16; off > 0; off >>= 1)
        lsum += __shfl_down(lsum, off, 32);
    if (lane == 0) wpart[wid] = lsum;
    __syncthreads();
    if (threadIdx.x == 0) {
        float s = 0.0f;
        for (int i = 0; i < 8; ++i) s += wpart[i];
        partials[blockIdx.x] = s;
    }
}

__global__ __launch_bounds__(256) void vq_reduce(const float* __restrict__ partials,
                                                 float* __restrict__ loss_out,
                                                 int nparts, float scale) {
    __shared__ float sh[256];
    float s = 0.0f;
    for (int i = threadIdx.x; i < nparts; i += 256) s += partials[i];
    sh[threadIdx.x] = s;
    __syncthreads();
    for (int off = 128; off > 0; off >>= 1) {
        if ((int)threadIdx.x < off) sh[threadIdx.x] += sh[threadIdx.x + off];
        __syncthreads();
    }
    if (threadIdx.x == 0) *loss_out = sh[0] * scale;
}

extern "C" void kernel_launch(void* const* d_in, const int* in_sizes, int n_in,
                              void* d_out, int out_size, void* d_ws, size_t ws_size,
                              hipStream_t stream) {
    const float* zin   = (const float*)d_in[0];   // [8,3,256,256] f32
    const float* table = (const float*)d_in[1];   // [512,3] f32
    float* outp     = (float*)d_out;              // [8,3,256,256] f32 ++ [1] loss
    float* partials = (float*)d_ws;               // NBLOCKS floats

    vq_main<<<NBLOCKS, 256, 0, stream>>>(zin, table, outp, partials);

    // loss = 11 * mean((z_q - z)^2) over 3*NPIX elements
    const float scale = 11.0f / (float)(3 * NPIX);
    vq_reduce<<<1, 256, 0, stream>>>(partials, outp + (out_size - 1), NBLOCKS, scale);
}